// CPCCLoss_43001212567639
// MI455X (gfx1250) — compile-verified
//
#include <hip/hip_runtime.h>
#include <math.h>

#define C_FINE   100
#define C_MID    20
#define C_COARSE 5
#define DIMS     128
#define EPSF     1e-12f

// padded proto row counts (multiples of 16 for WMMA tiles)
#define PF_ROWS 112   // 7 tiles
#define PM_ROWS 32    // 2 tiles
#define PC_ROWS 16    // 1 tile

typedef __attribute__((ext_vector_type(2))) float v2f;
typedef __attribute__((ext_vector_type(8))) float v8f;

// ---- workspace layout (in floats) ----
// [0            , 12800)            fine_sums   (100*128 f32)
// [12800        , 12928)            fine_counts (100 int, padded to 128)
// [12928        , 12928+14336)      Pf (112*128)
// [27264        , 27264+ 4096)      Pm ( 32*128)
// [31360        , 31360+ 2048)      Pc ( 16*128)
// [33408        , 33408+12544)      Gf (112*112)
// [45952        , 45952+ 1024)      Gm ( 32* 32)
// [46976        , 46976+  256)      Gc ( 16* 16)
#define WS_SUM   0
#define WS_CNT   12800
#define WS_PF    12928
#define WS_PM    27264
#define WS_PC    31360
#define WS_GF    33408
#define WS_GM    45952
#define WS_GC    46976
#define WS_ZERO_COUNT 33408   // zero everything up through Pc each launch

// -------------------------------------------------------------------------
__global__ void cpcc_zero_kernel(float* ws, int n) {
    int i = blockIdx.x * blockDim.x + threadIdx.x;
    if (i < n) ws[i] = 0.0f;
}

// -------------------------------------------------------------------------
// Streaming segment-sum: each block accumulates its row range into LDS
// (ds_add_f32), then flushes once to global accumulators (global_atomic_add_f32).
__global__ void cpcc_accum_kernel(const float* __restrict__ rep,
                                  const int* __restrict__ tgt,
                                  int N, float* __restrict__ gsum,
                                  int* __restrict__ gcnt) {
    __shared__ float ssum[C_FINE * DIMS];   // 51200 B
    __shared__ int   scnt[C_FINE];

    const int tid = threadIdx.x;            // 256 threads
    for (int i = tid; i < C_FINE * DIMS; i += blockDim.x) ssum[i] = 0.0f;
    if (tid < C_FINE) scnt[tid] = 0;
    __syncthreads();

    const int rowsPerBlk = (N + gridDim.x - 1) / gridDim.x;
    const int start = blockIdx.x * rowsPerBlk;
    const int end   = min(start + rowsPerBlk, N);

    const int d    = tid & (DIMS - 1);      // 0..127
    const int half = tid >> 7;              // 0 or 1: two rows per iteration

    for (int base = start; base < end; base += 2) {
        const int row = base + half;
        if (row < end) {
            const int cls = tgt[row];
            atomicAdd(&ssum[cls * DIMS + d], rep[(size_t)row * DIMS + d]);
            if (d == 0) atomicAdd(&scnt[cls], 1);
        }
    }
    __syncthreads();

    for (int i = tid; i < C_FINE * DIMS; i += blockDim.x)
        atomicAdd(&gsum[i], ssum[i]);
    if (tid < C_FINE) atomicAdd(&gcnt[tid], scnt[tid]);
}

// -------------------------------------------------------------------------
// Build fine/mid/coarse prototypes into zero-padded matrices. One block,
// 128 threads (thread = dimension). Total work is ~100*20 iterations: trivial.
__global__ void cpcc_protos_kernel(const float* __restrict__ gsum,
                                   const int* __restrict__ gcnt,
                                   const int* __restrict__ f2m,
                                   const int* __restrict__ f2c,
                                   float* __restrict__ Pf,
                                   float* __restrict__ Pm,
                                   float* __restrict__ Pc) {
    __shared__ int s_m2c[C_MID];
    const int d = threadIdx.x;

    if (d == 0) {
        for (int m = 0; m < C_MID; ++m) s_m2c[m] = -0x40000000;
        for (int f = 0; f < C_FINE; ++f) {   // segment_max(fine2coarse, fine2mid)
            const int m = f2m[f], c = f2c[f];
            if (c > s_m2c[m]) s_m2c[m] = c;
        }
    }
    __syncthreads();

    // fine prototypes = sums / max(count, 1)
    for (int f = 0; f < C_FINE; ++f) {
        const float c = fmaxf((float)gcnt[f], 1.0f);
        Pf[f * DIMS + d] = gsum[f * DIMS + d] / c;
    }
    // mid prototypes = mean of child fine protos
    for (int m = 0; m < C_MID; ++m) {
        float s = 0.0f; int c = 0;
        for (int f = 0; f < C_FINE; ++f)
            if (f2m[f] == m) { s += Pf[f * DIMS + d]; ++c; }
        Pm[m * DIMS + d] = s / fmaxf((float)c, 1.0f);
    }
    // coarse prototypes = mean of child mid protos (via mid2coarse)
    for (int cc = 0; cc < C_COARSE; ++cc) {
        float s = 0.0f; int c = 0;
        for (int m = 0; m < C_MID; ++m)
            if (s_m2c[m] == cc) { s += Pm[m * DIMS + d]; ++c; }
        Pc[cc * DIMS + d] = s / fmaxf((float)c, 1.0f);
    }
}

// -------------------------------------------------------------------------
// G = P * P^T via V_WMMA_F32_16X16X4_F32. One wave per block, one 16x16 tile
// per wave, K swept 128/4 = 32 WMMA issues per tile.
// A 16x4 f32 layout: lanes 0-15 carry M rows with K0,K1; lanes 16-31 K2,K3.
// B 4x16 mirrors with N in place of M, so for P*P^T both loads are row loads.
__global__ void cpcc_gram_wmma_kernel(const float* __restrict__ P,
                                      float* __restrict__ G, int tiles) {
    const int ti   = blockIdx.x / tiles;
    const int tj   = blockIdx.x % tiles;
    const int lane = threadIdx.x;
    const int half = lane >> 4;      // 0: K pair {0,1}; 1: K pair {2,3}
    const int l    = lane & 15;

    const float* rowA = P + (size_t)(ti * 16 + l) * DIMS;
    const float* rowB = P + (size_t)(tj * 16 + l) * DIMS;

    v8f c = {};
#pragma unroll
    for (int k0 = 0; k0 < DIMS; k0 += 4) {
        v2f a, b;
        a[0] = rowA[k0 + 2 * half];
        a[1] = rowA[k0 + 2 * half + 1];
        b[0] = rowB[k0 + 2 * half];
        b[1] = rowB[k0 + 2 * half + 1];
        c = __builtin_amdgcn_wmma_f32_16x16x4_f32(
                /*neg_a=*/false, a, /*neg_b=*/false, b,
                /*c_mod=*/(short)0, c, /*reuse_a=*/false, /*reuse_b=*/false);
    }

    const int ldG = tiles * 16;
    const int col = tj * 16 + l;
#pragma unroll
    for (int r = 0; r < 8; ++r) {
        const int row = ti * 16 + r + 8 * half;  // VGPR r: M=r (lanes 0-15), M=8+r (16-31)
        G[row * ldG + col] = c[r];
    }
}

// -------------------------------------------------------------------------
// 4950 pair distances + Pearson correlation, single block.
// dist^2(i,j) = G[i][i] + G[j][j] - 2 G[i][j]; safe_norm adds EPS inside sqrt.
__global__ void cpcc_corr_kernel(const int* __restrict__ f2m,
                                 const int* __restrict__ f2c,
                                 const float* __restrict__ Gf,
                                 const float* __restrict__ Gm,
                                 const float* __restrict__ Gc,
                                 float* __restrict__ out) {
    const int tid = threadIdx.x;   // 256 threads
    float st = 0.f, sp = 0.f, stp = 0.f, stt = 0.f, spp = 0.f;

    int p = 0;
    for (int i = 0; i < C_FINE - 1; ++i) {
        for (int j = i + 1; j < C_FINE; ++j) {
            if ((p & 255) == tid) {
                const int mi = f2m[i], mj = f2m[j];
                const int ci = f2c[i], cj = f2c[j];
                float q;
                q = Gf[i * PF_ROWS + i] + Gf[j * PF_ROWS + j] - 2.f * Gf[i * PF_ROWS + j];
                const float dp = sqrtf(fmaxf(q, 0.f) + EPSF);
                q = Gm[mi * PM_ROWS + mi] + Gm[mj * PM_ROWS + mj] - 2.f * Gm[mi * PM_ROWS + mj];
                const float dm = sqrtf(fmaxf(q, 0.f) + EPSF);
                q = Gc[ci * PC_ROWS + ci] + Gc[cj * PC_ROWS + cj] - 2.f * Gc[ci * PC_ROWS + cj];
                const float dc = sqrtf(fmaxf(q, 0.f) + EPSF);
                const float t = dc + dm;
                st += t;  sp += dp;
                stp += t * dp;  stt += t * t;  spp += dp * dp;
            }
            ++p;
        }
    }

    __shared__ float red[256];
    float acc[5] = { st, sp, stp, stt, spp };
#pragma unroll
    for (int k = 0; k < 5; ++k) {
        red[tid] = acc[k];
        __syncthreads();
        for (int s = 128; s > 0; s >>= 1) {
            if (tid < s) red[tid] += red[tid + s];
            __syncthreads();
        }
        acc[k] = red[0];
        __syncthreads();
    }

    if (tid == 0) {
        const float n   = (float)(C_FINE * (C_FINE - 1) / 2);
        const float cov = acc[2] - acc[0] * acc[1] / n;   // sum(tc*pc)
        const float vt  = acc[3] - acc[0] * acc[0] / n;   // sum(tc*tc)
        const float vp  = acc[4] - acc[1] * acc[1] / n;   // sum(pc*pc)
        const float corr = cov / sqrtf(vt * vp + EPSF);
        out[0] = 1.0f - corr;
    }
}

// -------------------------------------------------------------------------
extern "C" void kernel_launch(void* const* d_in, const int* in_sizes, int n_in,
                              void* d_out, int out_size, void* d_ws, size_t ws_size,
                              hipStream_t stream) {
    const float* rep = (const float*)d_in[0];
    const int*   tgt = (const int*)d_in[1];
    const int*   f2m = (const int*)d_in[2];
    const int*   f2c = (const int*)d_in[3];
    const int N = in_sizes[1];

    float* ws   = (float*)d_ws;
    float* gsum = ws + WS_SUM;
    int*   gcnt = (int*)(ws + WS_CNT);
    float* Pf   = ws + WS_PF;
    float* Pm   = ws + WS_PM;
    float* Pc   = ws + WS_PC;
    float* Gf   = ws + WS_GF;
    float* Gm   = ws + WS_GM;
    float* Gc   = ws + WS_GC;

    // 1) zero all accumulators + padded proto matrices (ws is poisoned)
    cpcc_zero_kernel<<<(WS_ZERO_COUNT + 255) / 256, 256, 0, stream>>>(ws, WS_ZERO_COUNT);

    // 2) streaming segment sum (the bandwidth-bound part: 128 MiB read)
    cpcc_accum_kernel<<<128, 256, 0, stream>>>(rep, tgt, N, gsum, gcnt);

    // 3) prototype tree (tiny)
    cpcc_protos_kernel<<<1, 128, 0, stream>>>(gsum, gcnt, f2m, f2c, Pf, Pm, Pc);

    // 4) Gram matrices via f32 WMMA (one wave = one 16x16 tile)
    cpcc_gram_wmma_kernel<<<(PF_ROWS / 16) * (PF_ROWS / 16), 32, 0, stream>>>(Pf, Gf, PF_ROWS / 16);
    cpcc_gram_wmma_kernel<<<(PM_ROWS / 16) * (PM_ROWS / 16), 32, 0, stream>>>(Pm, Gm, PM_ROWS / 16);
    cpcc_gram_wmma_kernel<<<1, 32, 0, stream>>>(Pc, Gc, PC_ROWS / 16);

    // 5) pairwise distances + Pearson correlation
    cpcc_corr_kernel<<<1, 256, 0, stream>>>(f2m, f2c, Gf, Gm, Gc, (float*)d_out);
}